// SP_loss_84146999263826
// MI455X (gfx1250) — compile-verified
//
#include <hip/hip_runtime.h>
#include <hip/hip_bf16.h>

// SP-loss for x: (8192, 256) fp32 row-normalized, T = 0.04, PIC_NUM = 4.
// s = x@x^T/T is never materialized: each 16-row panel (4 groups) streams
// 16x16 WMMA tiles of s and reduces sum(exp(s)) on the fly.
// v2: 4 independent column tiles per wave -> A fragment amortized over 4
// wmmas (1.25 loads/wmma instead of 2) and 4 concurrent accumulator chains.

typedef __attribute__((ext_vector_type(2))) float v2f;
typedef __attribute__((ext_vector_type(8))) float v8f;

#define NROWS 8192
#define DDIM  256
#define TVAL  0.04f
#define TINV  25.0f
#define NPANEL (NROWS / 16)   // 512 row panels
#define NCHUNK 16             // 16 chunks x 8 waves x 4 tiles = 512 col tiles

__global__ __launch_bounds__(256)
void sp_panel_kernel(const float* __restrict__ x, float* __restrict__ panel_out) {
  const int p    = blockIdx.x;        // panel: rows [16p, 16p+16)
  const int R0   = p * 16;
  const int tid  = threadIdx.x;
  const int wave = tid >> 5;
  const int lane = tid & 31;
  const int half = lane >> 4;         // 0: rows 0..7 / K 0,1   1: rows 8..15 / K 2,3
  const int lcol = lane & 15;

  __shared__ float part[8][16][16];   // [wave][row][col-residue] neg partials
  __shared__ float pose[4][4][4];     // exp(-pos) per group
  __shared__ float rowneg[16];
  __shared__ float terms[4];

  // per-lane streaming accumulators for sum(exp(s)) over non-group columns
  float acc[8];
#pragma unroll
  for (int j = 0; j < 8; ++j) acc[j] = 0.f;

  // A fragment address: row = R0 + lane%16, K pair = 2*half (contiguous float2)
  const float* pA = x + (size_t)(R0 + lcol) * DDIM + 2 * half;

  // streaming-LSE epilogue for one finished 16x16 tile
  auto accum_tile = [&](const v8f& c, int t) {
    const bool diag = (t == p);       // tile holding this panel's own groups
#pragma unroll
    for (int j = 0; j < 8; ++j) {
      const int  m      = j + 8 * half;                // row within panel
      const bool masked = diag && ((m >> 2) == (lcol >> 2));
      const float e = __expf(c[j] * TINV);             // exp(s), s in [-25,25]
      acc[j] += masked ? 0.f : e;
    }
  };

  for (int ch = 0; ch < NCHUNK; ++ch) {
    const int t0 = (ch * 8 + wave) * 4;                // 4 consecutive tiles
    const float* pB = x + (size_t)(t0 * 16 + lcol) * DDIM + 2 * half;
    if (ch + 1 < NCHUNK) {
      // next chunk for this wave is +32 tiles = +512 rows ahead
      __builtin_prefetch((const void*)(pB + (size_t)512 * DDIM), 0, 0);
    }

    v8f c0 = {0.f,0.f,0.f,0.f,0.f,0.f,0.f,0.f};
    v8f c1 = c0, c2 = c0, c3 = c0;

#pragma unroll 8
    for (int k = 0; k < DDIM; k += 4) {
      v2f a  = *(const v2f*)(pA + k);                  // shared A fragment
      v2f b0 = *(const v2f*)(pB + k);
      v2f b1 = *(const v2f*)(pB + (size_t)16 * DDIM + k);
      v2f b2 = *(const v2f*)(pB + (size_t)32 * DDIM + k);
      v2f b3 = *(const v2f*)(pB + (size_t)48 * DDIM + k);
      c0 = __builtin_amdgcn_wmma_f32_16x16x4_f32(false, a, false, b0, (short)0, c0, false, false);
      c1 = __builtin_amdgcn_wmma_f32_16x16x4_f32(false, a, false, b1, (short)0, c1, false, false);
      c2 = __builtin_amdgcn_wmma_f32_16x16x4_f32(false, a, false, b2, (short)0, c2, false, false);
      c3 = __builtin_amdgcn_wmma_f32_16x16x4_f32(false, a, false, b3, (short)0, c3, false, false);
    }

    accum_tile(c0, t0 + 0);
    accum_tile(c1, t0 + 1);
    accum_tile(c2, t0 + 2);
    accum_tile(c3, t0 + 3);
  }

  // spill per-lane partials (unique slot per (wave,row,col-residue): no atomics)
#pragma unroll
  for (int j = 0; j < 8; ++j)
    part[wave][j + 8 * half][lcol] = acc[j];

  __syncthreads();

  // deterministic fixed-order row reduction: rowneg[r] = sum over 8 waves x 16 cols
  if (tid < 16) {
    float s = 0.f;
    for (int w = 0; w < 8; ++w)
      for (int cc = 0; cc < 16; ++cc)
        s += part[w][tid][cc];
    rowneg[tid] = s;
  }

  // pos blocks: 64 threads, one (group q, i, j) dot each (tiny: 4*16*256 FMA)
  if (tid < 64) {
    const int q  = tid >> 4;
    const int e  = tid & 15;
    const int i  = e >> 2;
    const int jj = e & 3;
    const float* xr = x + (size_t)(R0 + 4 * q + i)  * DDIM;
    const float* xc = x + (size_t)(R0 + 4 * q + jj) * DDIM;
    float d = 0.f;
    for (int k = 0; k < DDIM; ++k) d = fmaf(xr[k], xc[k], d);
    pose[q][i][jj] = __expf(-d * TINV);
  }
  __syncthreads();

  // per-group scalar epilogue
  if (tid < 4) {
    const int q = tid;
    float Hsum = 0.f, Lsum = 0.f;
#pragma unroll
    for (int i = 0; i < 4; ++i) {
      const float rs = pose[q][i][0] + pose[q][i][1] + pose[q][i][2] + pose[q][i][3];
      Hsum += rs;            // sum exp(-pos) over 4x4
      Lsum += 1.0f / rs;     // sum_i exp(-lse_row_i)
    }
    const float hard_pos  = -TVAL * logf(Hsum);
    const float least_pos =  TVAL * logf(Lsum);
    const float negsum = rowneg[4*q] + rowneg[4*q+1] + rowneg[4*q+2] + rowneg[4*q+3];
    const float neg = TVAL * logf(negsum);
    const float alpha = (hard_pos >= 0.f)
        ? (2.0f * least_pos * hard_pos / (hard_pos + least_pos)) : 0.f;
    const float margin = alpha * hard_pos + (1.0f - alpha) * least_pos;
    const float z  = (neg - margin) * TINV;
    const float sp = fmaxf(z, 0.f) + log1pf(__expf(-fabsf(z)));  // stable softplus
    terms[q] = sp;
  }
  __syncthreads();

  if (tid == 0)
    panel_out[p] = terms[0] + terms[1] + terms[2] + terms[3];
}

__global__ __launch_bounds__(256)
void sp_reduce_kernel(const float* __restrict__ panel_out, float* __restrict__ out) {
  __shared__ float s[256];
  const int tid = threadIdx.x;
  s[tid] = panel_out[tid] + panel_out[tid + 256];
  __syncthreads();
  for (int ofs = 128; ofs > 0; ofs >>= 1) {   // fixed pairing: deterministic
    if (tid < ofs) s[tid] += s[tid + ofs];
    __syncthreads();
  }
  if (tid == 0) out[0] = s[0] * (1.0f / 2048.0f);   // mean over G = 2048 groups
}

extern "C" void kernel_launch(void* const* d_in, const int* in_sizes, int n_in,
                              void* d_out, int out_size, void* d_ws, size_t ws_size,
                              hipStream_t stream) {
  const float* x  = (const float*)d_in[0];
  float* ws  = (float*)d_ws;    // NPANEL floats of scratch
  float* out = (float*)d_out;   // scalar loss

  sp_panel_kernel<<<NPANEL, 256, 0, stream>>>(x, ws);
  sp_reduce_kernel<<<1, 256, 0, stream>>>(ws, out);
}